// GCN_34376918237986
// MI455X (gfx1250) — compile-verified
//
#include <hip/hip_runtime.h>

typedef __attribute__((ext_vector_type(2))) float v2f;
typedef __attribute__((ext_vector_type(8))) float v8f;

#define GCN_N 40000
#define GCN_E 640000

// ---------------- degree / norm ----------------
__global__ void gcn_deg_kernel(const int* __restrict__ src, const int* __restrict__ dst,
                               float* __restrict__ outdeg, float* __restrict__ indeg, int E) {
    int e = blockIdx.x * blockDim.x + threadIdx.x;
    if (e < E) {
        atomicAdd(&outdeg[src[e]], 1.0f);
        atomicAdd(&indeg[dst[e]], 1.0f);
    }
}

__global__ void gcn_norm_kernel(const float* __restrict__ outdeg, const float* __restrict__ indeg,
                                float* __restrict__ outnorm, float* __restrict__ innorm, int n) {
    int i = blockIdx.x * blockDim.x + threadIdx.x;
    if (i < n) {
        float od = outdeg[i], idg = indeg[i];
        outnorm[i] = od > 0.0f ? rsqrtf(fmaxf(od, 1.0f)) : 0.0f;
        innorm[i]  = idg > 0.0f ? rsqrtf(fmaxf(idg, 1.0f)) : 0.0f;
    }
}

// ---------------- WMMA fp32 GEMM: C[N x NC] = (pre ? diag(pre) : I) * A[N x K] @ W[K x NC]
//                  epilogue: if post: v = v*post[row] + bias[col];  if relu: max(v,0)
// one wave (32 lanes) per 16x16 output tile; N, NC multiples of 16; K multiple of 4.
__global__ __launch_bounds__(32)
void gcn_wmma_gemm_kernel(const float* __restrict__ A, const float* __restrict__ W,
                          float* __restrict__ C, int K, int NC,
                          const float* __restrict__ pre, const float* __restrict__ post,
                          const float* __restrict__ bias, int relu) {
    const int lane = threadIdx.x;        // 0..31
    const int half = lane >> 4;          // 0 or 1
    const int lr   = lane & 15;          // 0..15
    const int row0 = blockIdx.x * 16;
    const int col0 = blockIdx.y * 16;

    const float prescale = pre ? pre[row0 + lr] : 1.0f;
    const float* __restrict__ arow = A + (size_t)(row0 + lr) * K;

    v8f acc = {};
    for (int k0 = 0; k0 < K; k0 += 4) {
        const int ka = k0 + 2 * half;    // k index held by this half-wave, VGPR 0
        v2f a, b;
        // A fragment: 16x4, row m = lane%16; lanes0-15 hold k0,k0+1; lanes16-31 hold k0+2,k0+3
        a.x = arow[ka]     * prescale;
        a.y = arow[ka + 1] * prescale;
        // B fragment: 4x16, col n = lane%16; k split mirrored across half-wave/VGPR
        b.x = W[(size_t)ka       * NC + col0 + lr];
        b.y = W[(size_t)(ka + 1) * NC + col0 + lr];
        acc = __builtin_amdgcn_wmma_f32_16x16x4_f32(false, a, false, b, (short)0, acc,
                                                    false, false);
    }

    // C/D layout: VGPR v holds row m = v + 8*half, col n = lane%16
#pragma unroll
    for (int v = 0; v < 8; ++v) {
        const int m = row0 + v + 8 * half;
        const int n = col0 + lr;
        float val = acc[v];
        if (post) val = val * post[m] + bias[n];
        if (relu) val = fmaxf(val, 0.0f);
        C[(size_t)m * NC + n] = val;
    }
}

// ---------------- edge scatter: out[dst[e], :] += (pre ? pre[src[e]] : 1) * x[src[e], :]
// vectorized float4 per thread; shift = log2(dim/4)
__global__ void gcn_scatter_kernel(const float* __restrict__ x, const int* __restrict__ src,
                                   const int* __restrict__ dst, const float* __restrict__ pre,
                                   float* __restrict__ out, int dim, int shift, int E) {
    int tid = blockIdx.x * blockDim.x + threadIdx.x;
    int e = tid >> shift;
    if (e >= E) return;
    int f = (tid & ((1 << shift) - 1)) << 2;
    int s = src[e], d = dst[e];
    float scale = pre ? pre[s] : 1.0f;
    const float4 vx = *(const float4*)(x + (size_t)s * dim + f);
    float* o = out + (size_t)d * dim + f;
    atomicAdd(o + 0, vx.x * scale);
    atomicAdd(o + 1, vx.y * scale);
    atomicAdd(o + 2, vx.z * scale);
    atomicAdd(o + 3, vx.w * scale);
}

// ---------------- elementwise epilogue (for GEMM-first layers): h = relu(s*innorm[row] + bias[col])
__global__ void gcn_epilogue_kernel(const float* __restrict__ s, const float* __restrict__ innorm,
                                    const float* __restrict__ bias, float* __restrict__ h,
                                    int shiftD, int maskD, int total, int relu) {
    int i = blockIdx.x * blockDim.x + threadIdx.x;
    if (i < total) {
        int r = i >> shiftD;
        int c = i & maskD;
        float v = s[i] * innorm[r] + bias[c];
        if (relu) v = fmaxf(v, 0.0f);
        h[i] = v;
    }
}

// ---------------- layer 7 helpers (64 -> 1) ----------------
__global__ void gcn_rowdot_kernel(const float* __restrict__ x, const float* __restrict__ outnorm,
                                  const float* __restrict__ w, float* __restrict__ t,
                                  int n, int K) {
    int i = blockIdx.x * blockDim.x + threadIdx.x;
    if (i < n) {
        const float* xr = x + (size_t)i * K;
        float acc = 0.0f;
        for (int k = 0; k < K; ++k) acc = fmaf(xr[k], w[k], acc);
        t[i] = acc * outnorm[i];
    }
}

__global__ void gcn_scatter1_kernel(const float* __restrict__ t, const int* __restrict__ src,
                                    const int* __restrict__ dst, float* __restrict__ out, int E) {
    int e = blockIdx.x * blockDim.x + threadIdx.x;
    if (e < E) atomicAdd(&out[dst[e]], t[src[e]]);
}

__global__ void gcn_final_kernel(const float* __restrict__ s, const float* __restrict__ innorm,
                                 const float* __restrict__ b, float* __restrict__ out, int n) {
    int i = blockIdx.x * blockDim.x + threadIdx.x;
    if (i < n) out[i] = s[i] * innorm[i] + b[0];
}

extern "C" void kernel_launch(void* const* d_in, const int* in_sizes, int n_in,
                              void* d_out, int out_size, void* d_ws, size_t ws_size,
                              hipStream_t stream) {
    (void)in_sizes; (void)n_in; (void)out_size; (void)ws_size;

    const float* in_feat = (const float*)d_in[0];
    const int*   src     = (const int*)d_in[1];
    const int*   dst     = (const int*)d_in[2];
    const float* W[7];
    const float* B[7];
    for (int i = 0; i < 7; ++i) {
        W[i] = (const float*)d_in[3 + 2 * i];
        B[i] = (const float*)d_in[4 + 2 * i];
    }
    float* out = (float*)d_out;

    float* ws      = (float*)d_ws;
    float* outdeg  = ws;
    float* indeg   = ws + GCN_N;
    float* outnorm = ws + 2 * (size_t)GCN_N;
    float* innorm  = ws + 3 * (size_t)GCN_N;
    float* bufA    = ws + 4 * (size_t)GCN_N;
    float* bufB    = bufA + (size_t)GCN_N * 128;
    float* bufC    = bufB + (size_t)GCN_N * 128;

    // degrees + norms
    hipMemsetAsync(outdeg, 0, 2 * (size_t)GCN_N * sizeof(float), stream);
    gcn_deg_kernel<<<(GCN_E + 255) / 256, 256, 0, stream>>>(src, dst, outdeg, indeg, GCN_E);
    gcn_norm_kernel<<<(GCN_N + 255) / 256, 256, 0, stream>>>(outdeg, indeg, outnorm, innorm, GCN_N);

    auto gemm = [&](const float* A, const float* Wt, float* Cc, int K, int NC,
                    const float* pre, const float* post, const float* bias, int relu) {
        dim3 grid(GCN_N / 16, NC / 16);
        gcn_wmma_gemm_kernel<<<grid, 32, 0, stream>>>(A, Wt, Cc, K, NC, pre, post, bias, relu);
    };
    auto scat = [&](const float* x, float* o, int dim, const float* pre) {
        hipMemsetAsync(o, 0, (size_t)GCN_N * dim * sizeof(float), stream);
        int shift = (dim == 128) ? 5 : 4;            // float4 chunks per row
        long total = (long)GCN_E << shift;
        gcn_scatter_kernel<<<(int)((total + 255) / 256), 256, 0, stream>>>(
            x, src, dst, pre, o, dim, shift, GCN_E);
    };
    auto epi = [&](const float* s, const float* bias, float* h, int dim, int relu) {
        int shiftD = (dim == 128) ? 7 : 6;
        int total = GCN_N * dim;
        gcn_epilogue_kernel<<<(total + 255) / 256, 256, 0, stream>>>(
            s, innorm, bias, h, shiftD, dim - 1, total, relu);
    };

    // L1: 128->64, GEMM first (shrink message dim)
    gemm(in_feat, W[0], bufA, 128, 64, outnorm, nullptr, nullptr, 0);
    scat(bufA, bufB, 64, nullptr);
    epi(bufB, B[0], bufC, 64, 1);
    // L2: 64->128, scatter first
    scat(bufC, bufA, 64, outnorm);
    gemm(bufA, W[1], bufB, 64, 128, nullptr, innorm, B[1], 1);
    // L3: 128->128, scatter first
    scat(bufB, bufA, 128, outnorm);
    gemm(bufA, W[2], bufC, 128, 128, nullptr, innorm, B[2], 1);
    // L4: 128->64, GEMM first
    gemm(bufC, W[3], bufA, 128, 64, outnorm, nullptr, nullptr, 0);
    scat(bufA, bufB, 64, nullptr);
    epi(bufB, B[3], bufC, 64, 1);
    // L5: 64->64, scatter first
    scat(bufC, bufA, 64, outnorm);
    gemm(bufA, W[4], bufB, 64, 64, nullptr, innorm, B[4], 1);
    // L6: 64->64, scatter first
    scat(bufB, bufA, 64, outnorm);
    gemm(bufA, W[5], bufC, 64, 64, nullptr, innorm, B[5], 1);
    // L7: 64->1, GEMM (row-dot) first, then dim-1 scatter, final affine (no relu)
    gcn_rowdot_kernel<<<(GCN_N + 255) / 256, 256, 0, stream>>>(bufC, outnorm, W[6], bufA, GCN_N, 64);
    hipMemsetAsync(bufB, 0, (size_t)GCN_N * sizeof(float), stream);
    gcn_scatter1_kernel<<<(GCN_E + 255) / 256, 256, 0, stream>>>(bufA, src, dst, bufB, GCN_E);
    gcn_final_kernel<<<(GCN_N + 255) / 256, 256, 0, stream>>>(bufB, innorm, B[6], out, GCN_N);
}